// S4_43499428774023
// MI455X (gfx1250) — compile-verified
//
#include <hip/hip_runtime.h>
#include <hip/hip_bf16.h>

typedef float v2f __attribute__((ext_vector_type(2)));
typedef float v8f __attribute__((ext_vector_type(8)));
typedef int   v4i __attribute__((ext_vector_type(4)));
typedef __attribute__((address_space(1))) v4i gas_v4i;
typedef __attribute__((address_space(3))) v4i las_v4i;

#define BSZ   8
#define LSEQ  2048
#define DCH   256
#define NST   64
#define CHK   16
#define NCHUNK (LSEQ/CHK)
#define EPSV  1e-6f

// ---- workspace layout (float offsets) ----
#define OFS_A     0u          // [N,N]   softplus(log_A)
#define OFS_AINV  4096u       // [N,N]   (A+eps I)^-1
#define OFS_X     8192u       // [D,N]   X[d] = Ainv @ Bp[d]
#define OFS_DA    24576u      // [D,N,N] expm(A*dt_d)
#define OFS_DB    1073152u    // [D,N]
#define OFS_P     1089536u    // [D,N,N] dA^16
#define OFS_E     2138112u    // [D,N,16]
#define OFS_G     2400256u    // [D,16,N]
#define OFS_KT    2662400u    // [D,16,16]
#define OFS_XT    2727936u    // [D,B,L] transposed x
#define OFS_Y     6922240u    // [B,L,D] conv output

#if __has_builtin(__builtin_amdgcn_global_load_async_to_lds_b128) && \
    __has_builtin(__builtin_amdgcn_s_wait_asynccnt)
#define USE_ASYNC_LDS 1
#else
#define USE_ASYNC_LDS 0
#endif

__device__ __forceinline__ float softplus_f(float v) {
  return v > 20.f ? v : log1pf(expf(v));
}

// 64x64 fp32 matmul in LDS, blockDim.x == 256, caller handles barriers
__device__ __forceinline__ void mm64(float* dst, const float* a, const float* b, float scale) {
#pragma unroll
  for (int t = 0; t < 16; ++t) {
    int idx = threadIdx.x + t * 256;
    int r = idx >> 6, c = idx & 63;
    float s = 0.f;
    for (int k = 0; k < NST; ++k) s = fmaf(a[r * NST + k], b[k * NST + c], s);
    dst[idx] = s * scale;
  }
}

// ---------------- Stage 0a: A = softplus(log_A); Ainv = (A+eps I)^-1 ----------------
__global__ void s4_prep_inv(const float* __restrict__ logA, float* __restrict__ ws) {
  __shared__ float M[NST * NST];
  __shared__ float Inv[NST * NST];
  __shared__ float fcol[NST];
  int tid = threadIdx.x;
  for (int idx = tid; idx < NST * NST; idx += 256) {
    float v = softplus_f(logA[idx]);
    ws[OFS_A + idx] = v;
    int r = idx >> 6, c = idx & 63;
    M[idx] = v + (r == c ? EPSV : 0.f);
    Inv[idx] = (r == c) ? 1.f : 0.f;
  }
  __syncthreads();
  for (int p = 0; p < NST; ++p) {
    float ip = 1.f / M[p * NST + p];
    __syncthreads();
    if (tid < 2 * NST) {
      int mt = tid >> 6, j = tid & 63;
      if (mt == 0) M[p * NST + j] *= ip; else Inv[p * NST + j] *= ip;
    }
    __syncthreads();
    if (tid < NST) fcol[tid] = (tid == p) ? 0.f : M[tid * NST + p];
    __syncthreads();
    for (int idx = tid; idx < NST * NST; idx += 256) {
      int r = idx >> 6, j = idx & 63;
      if (r != p) {
        float f = fcol[r];
        M[idx]   -= f * M[p * NST + j];
        Inv[idx] -= f * Inv[p * NST + j];
      }
    }
    __syncthreads();
  }
  for (int idx = tid; idx < NST * NST; idx += 256) ws[OFS_AINV + idx] = Inv[idx];
}

// ---------------- Stage 0b: X[d] = Ainv @ Bp[d]  (grid=D, block=64) ----------------
__global__ void s4_prep_X(const float* __restrict__ Bp, const float* __restrict__ wsAinv,
                          float* __restrict__ wsX) {
  __shared__ float bv[NST];
  int d = blockIdx.x, n = threadIdx.x;
  bv[n] = Bp[d * NST + n];
  __syncthreads();
  float s = 0.f;
  for (int m = 0; m < NST; ++m) s = fmaf(wsAinv[n * NST + m], bv[m], s);
  wsX[d * NST + n] = s;
}

// ---------------- Stage 1: dA_d = expm(A*dt_d), dB_d = (dA-I)@X[d]  (grid=D, block=256) ----------------
__global__ void s4_discretize(const float* __restrict__ log_delta, const float* __restrict__ wsA,
                              const float* __restrict__ wsX, float* __restrict__ wsDA,
                              float* __restrict__ wsDB) {
  __shared__ float lT[NST * NST];
  __shared__ float lU[NST * NST];
  __shared__ float lV[NST * NST];
  __shared__ float xv[NST];
  int d = blockIdx.x, tid = threadIdx.x;
  float dt = softplus_f(log_delta[d]) + 1e-6f;
  float c4 = dt * (1.f / 16.f);                // scaling: 2^-4
  float rAcc[16];
#pragma unroll
  for (int t = 0; t < 16; ++t) {
    int idx = tid + t * 256;
    float v = wsA[idx] * c4;
    lT[idx] = v; lU[idx] = v;
    int r = idx >> 6, c = idx & 63;
    rAcc[t] = (r == c ? 1.f : 0.f) + v;        // I + T
  }
  __syncthreads();
  float* cu = lU; float* cv = lV;
  for (int j = 2; j <= 8; ++j) {               // Taylor terms 2..8
    mm64(cv, cu, lT, 1.f / (float)j);
    __syncthreads();
#pragma unroll
    for (int t = 0; t < 16; ++t) rAcc[t] += cv[tid + t * 256];
    float* tmp = cu; cu = cv; cv = tmp;
    __syncthreads();
  }
#pragma unroll
  for (int t = 0; t < 16; ++t) cu[tid + t * 256] = rAcc[t];
  __syncthreads();
  for (int sq = 0; sq < 4; ++sq) {             // squaring: ^16
    mm64(cv, cu, cu, 1.f);
    __syncthreads();
    float* tmp = cu; cu = cv; cv = tmp;
  }
  for (int idx = tid; idx < NST * NST; idx += 256)
    wsDA[(size_t)d * NST * NST + idx] = cu[idx];
  if (tid < NST) xv[tid] = wsX[d * NST + tid];
  __syncthreads();
  if (tid < NST) {
    float s = 0.f;
    for (int m = 0; m < NST; ++m) s = fmaf(cu[tid * NST + m], xv[m], s);
    wsDB[d * NST + tid] = s - xv[tid];
  }
}

// ---------------- Stage 2: chunk operators P,E,G,Ktri  (grid=D, block=256) ----------------
__global__ void s4_chunkops(const float* __restrict__ Cp, const float* __restrict__ wsDA,
                            const float* __restrict__ wsDB, float* __restrict__ wsP,
                            float* __restrict__ wsE, float* __restrict__ wsG,
                            float* __restrict__ wsKT) {
  __shared__ float la[NST * NST];
  __shared__ float p0[NST * NST];
  __shared__ float p1[NST * NST];
  __shared__ float v0[NST], v1[NST], vc[NST], vb[NST], kin[CHK];
  int d = blockIdx.x, tid = threadIdx.x;
  const float* dA = wsDA + (size_t)d * NST * NST;
  for (int idx = tid; idx < NST * NST; idx += 256) { la[idx] = dA[idx]; p0[idx] = dA[idx]; }
  if (tid < NST) { vb[tid] = wsDB[d * NST + tid]; vc[tid] = Cp[d * NST + tid]; v0[tid] = vb[tid]; }
  __syncthreads();
  // P = dA^16
  mm64(p1, p0, p0, 1.f); __syncthreads();
  mm64(p0, p1, p1, 1.f); __syncthreads();
  mm64(p1, p0, p0, 1.f); __syncthreads();
  mm64(p0, p1, p1, 1.f); __syncthreads();
  for (int idx = tid; idx < NST * NST; idx += 256) wsP[(size_t)d * NST * NST + idx] = p0[idx];
  // E[:,j] = dA^{15-j} dB  (row-major [n][j])
  if (tid < NST) wsE[(size_t)d * NST * CHK + tid * CHK + 15] = v0[tid];
  __syncthreads();
  for (int j = 14; j >= 0; --j) {
    if (tid < NST) {
      float s = 0.f;
      for (int m = 0; m < NST; ++m) s = fmaf(la[tid * NST + m], v0[m], s);
      v1[tid] = s;
    }
    __syncthreads();
    if (tid < NST) { v0[tid] = v1[tid]; wsE[(size_t)d * NST * CHK + tid * CHK + j] = v1[tid]; }
    __syncthreads();
  }
  // G[t] = C^T dA^{t+1}  (row-major [t][n])
  if (tid < NST) v0[tid] = vc[tid];
  __syncthreads();
  for (int t = 0; t < CHK; ++t) {
    if (tid < NST) {
      float s = 0.f;
      for (int n = 0; n < NST; ++n) s = fmaf(v0[n], la[n * NST + tid], s);
      v1[tid] = s;
    }
    __syncthreads();
    if (tid < NST) { v0[tid] = v1[tid]; wsG[(size_t)d * CHK * NST + t * NST + tid] = v1[tid]; }
    __syncthreads();
  }
  // intra-chunk taps k[tau] = C^T dA^tau dB
  if (tid < CHK) {
    float s = 0.f;
    if (tid == 0) { for (int n = 0; n < NST; ++n) s = fmaf(vc[n], vb[n], s); }
    else { for (int n = 0; n < NST; ++n) s = fmaf(wsG[(size_t)d * CHK * NST + (tid - 1) * NST + n], vb[n], s); }
    kin[tid] = s;
  }
  __syncthreads();
  if (tid < CHK * CHK) {
    int t = tid >> 4, j = tid & 15;
    wsKT[(size_t)d * CHK * CHK + tid] = (j <= t) ? kin[t - j] : 0.f;
  }
}

// ---------------- Stage 3: transpose x[B,L,D] -> xT[D,B,L] ----------------
__global__ void s4_transpose_x(const float* __restrict__ x, float* __restrict__ xT) {
  __shared__ float tile[32][33];
  int r0 = blockIdx.x * 32, d0 = blockIdx.y * 32;
  int tx = threadIdx.x, ty = threadIdx.y;   // 32 x 8
#pragma unroll
  for (int q = 0; q < 4; ++q)
    tile[ty + q * 8][tx] = x[(size_t)(r0 + ty + q * 8) * DCH + d0 + tx];
  __syncthreads();
#pragma unroll
  for (int q = 0; q < 4; ++q)
    xT[(size_t)(d0 + ty + q * 8) * (BSZ * LSEQ) + r0 + tx] = tile[tx][ty + q * 8];
}

// ---------------- WMMA fragment helpers (V_WMMA_F32_16X16X4_F32 layouts) ----------------
// A (16x4 f32): lane L: m=L&15, half=L>>4; holds A[m][k0+2*half], A[m][k0+2*half+1]
__device__ __forceinline__ v2f fragA(const float* base, int ld, int row0, int k0, int lane) {
  int m = lane & 15, half = lane >> 4;
  const float* p = base + (row0 + m) * ld + k0 + half * 2;
  v2f a; a.x = p[0]; a.y = p[1];
  return a;
}
// B (4x16 f32) from row-major [k][n] storage
__device__ __forceinline__ v2f fragB(const float* base, int ld, int k0, int col0, int lane) {
  int n = lane & 15, half = lane >> 4;
  const float* p = base + (k0 + half * 2) * ld + col0 + n;
  v2f b; b.x = p[0]; b.y = p[ld];
  return b;
}
// B (4x16 f32) from TRANSPOSED storage [n][k] (contiguous pair -> one ds_load_b64)
__device__ __forceinline__ v2f fragBT(const float* base, int ld, int k0, int col0, int lane) {
  int n = lane & 15, half = lane >> 4;
  const float* p = base + (col0 + n) * ld + k0 + half * 2;
  v2f b; b.x = p[0]; b.y = p[1];
  return b;
}
#define WMMA_F32(a, b, c) \
  __builtin_amdgcn_wmma_f32_16x16x4_f32(false, (a), false, (b), (short)0, (c), false, false)

// ---------------- Stage 4: chunked WMMA scan (grid=D, one wave per channel) ----------------
// S(64x16) <- P(64x64)@S + E(64x16)@X(16x16);  Y(16x16) = G(16x64)@S + Ktri(16x16)@X
__global__ void __launch_bounds__(32) s4_scan(const float* __restrict__ xT,
                                              const float* __restrict__ wsP,
                                              const float* __restrict__ wsE,
                                              const float* __restrict__ wsG,
                                              const float* __restrict__ wsKT,
                                              float* __restrict__ y) {
  __shared__ float sP[NST * NST];
  __shared__ float sE[NST * CHK];
  __shared__ float sG[CHK * NST];
  __shared__ float sK[CHK * CHK];
  __shared__ float sS[NST * CHK];
  __shared__ __align__(16) float sXc[2][CHK * CHK];   // batch-major: [n][t], double buffered
  int d = blockIdx.x, lane = threadIdx.x;
  for (int i = lane; i < NST * NST; i += 32) sP[i] = wsP[(size_t)d * NST * NST + i];
  for (int i = lane; i < NST * CHK; i += 32) { sE[i] = wsE[(size_t)d * NST * CHK + i]; sS[i] = 0.f; }
  for (int i = lane; i < CHK * NST; i += 32) sG[i] = wsG[(size_t)d * CHK * NST + i];
  for (int i = lane; i < CHK * CHK; i += 32) sK[i] = wsKT[(size_t)d * CHK * CHK + i];
  for (int i = lane; i < 2 * CHK * CHK; i += 32) (&sXc[0][0])[i] = 0.f;
  __syncthreads();

  // hoist chunk-invariant A-fragments into registers (72 VGPRs)
  v2f aE[4][4], aG[16], aK[4];
#pragma unroll
  for (int i = 0; i < 4; ++i)
#pragma unroll
    for (int kb = 0; kb < 4; ++kb) aE[i][kb] = fragA(sE, CHK, i * 16, kb * 4, lane);
#pragma unroll
  for (int kb = 0; kb < 16; ++kb) aG[kb] = fragA(sG, NST, 0, kb * 4, lane);
#pragma unroll
  for (int kb = 0; kb < 4; ++kb) aK[kb] = fragA(sK, CHK, 0, kb * 4, lane);

  int bb = lane >> 2, tq = (lane & 3) * 4;                 // lane -> (batch, time quad)
  const float* xbase = xT + ((size_t)d * BSZ + bb) * LSEQ + tq;
  int m = lane & 15, half = lane >> 4;

#if USE_ASYNC_LDS
  // gfx1250 async global->LDS copy: each lane moves 16 contiguous bytes, ASYNCcnt-tracked
#define SCAN_ISSUE_X(buf, toff)                                                  \
  __builtin_amdgcn_global_load_async_to_lds_b128(                                \
      (gas_v4i*)(xbase + (toff)),                                                \
      (las_v4i*)&sXc[(buf)][bb * CHK + tq], 0, 0)
#define SCAN_WAIT_X() __builtin_amdgcn_s_wait_asynccnt(0)
#else
#define SCAN_ISSUE_X(buf, toff)                                                  \
  do {                                                                           \
    float4 xv_ = *(const float4*)(xbase + (toff));                               \
    *(float4*)&sXc[(buf)][bb * CHK + tq] = xv_;                                  \
  } while (0)
#define SCAN_WAIT_X() do { } while (0)
#endif

  SCAN_ISSUE_X(0, 0);
  SCAN_WAIT_X();
  __syncthreads();
  int cur = 0;
  for (int c = 0; c < NCHUNK; ++c) {
    int t0 = c * CHK;
    if (c + 1 < NCHUNK) {
#if !USE_ASYNC_LDS
      __builtin_prefetch(xbase + t0 + CHK, 0, 1);          // global_prefetch_b8
#endif
      SCAN_ISSUE_X(cur ^ 1, t0 + CHK);                     // overlap copy with compute
    }
    const float* sX = sXc[cur];
    v8f zero = {0.f, 0.f, 0.f, 0.f, 0.f, 0.f, 0.f, 0.f};
    v8f accS[4] = {zero, zero, zero, zero};
    v8f accY = zero;
#pragma unroll
    for (int i = 0; i < 4; ++i) {
#pragma unroll
      for (int kb = 0; kb < 16; ++kb) {
        v2f a = fragA(sP, NST, i * 16, kb * 4, lane);
        v2f b = fragB(sS, CHK, kb * 4, 0, lane);
        accS[i] = WMMA_F32(a, b, accS[i]);
      }
#pragma unroll
      for (int kb = 0; kb < 4; ++kb) {
        v2f b = fragBT(sX, CHK, kb * 4, 0, lane);
        accS[i] = WMMA_F32(aE[i][kb], b, accS[i]);
      }
    }
#pragma unroll
    for (int kb = 0; kb < 16; ++kb) {
      v2f b = fragB(sS, CHK, kb * 4, 0, lane);
      accY = WMMA_F32(aG[kb], b, accY);
    }
#pragma unroll
    for (int kb = 0; kb < 4; ++kb) {
      v2f b = fragBT(sX, CHK, kb * 4, 0, lane);
      accY = WMMA_F32(aK[kb], b, accY);
    }
    __syncthreads();
    // write back state (C-layout: row = r + 8*half, col = m)
#pragma unroll
    for (int i = 0; i < 4; ++i)
#pragma unroll
      for (int r = 0; r < 8; ++r)
        sS[(i * 16 + r + half * 8) * CHK + m] = accS[i][r];
    // emit y[b, t0+t', d] for valid batches (col m = batch)
    if (m < BSZ) {
      float* yp = y + ((size_t)m * LSEQ + t0 + half * 8) * DCH + d;
#pragma unroll
      for (int r = 0; r < 8; ++r) yp[(size_t)r * DCH] = accY[r];
    }
    SCAN_WAIT_X();   // next chunk's async copy complete before it is consumed
    __syncthreads();
    cur ^= 1;
  }
}

// ---------------- Stage 5: out = (y + x*skip) @ W^T + b   (M=16384, N=256, K=256) ----------------
__global__ void __launch_bounds__(128) s4_outgemm(const float* __restrict__ x,
                                                  const float* __restrict__ yws,
                                                  const float* __restrict__ skipD,
                                                  const float* __restrict__ W,
                                                  const float* __restrict__ bout,
                                                  float* __restrict__ out) {
  __shared__ float sA[16 * 64];      // A k-slab: 16 rows x 64 k
  __shared__ float sW[64 * 65];      // B k-slab: 64 k x 64 n (padded stride 65)
  __shared__ float sBias[64];
  int m0 = blockIdx.x * 16, n0 = blockIdx.y * 64;
  int tid = threadIdx.x, lane = tid & 31, w = tid >> 5;
  int m = lane & 15, half = lane >> 4;
  if (tid < 64) sBias[tid] = bout[n0 + tid];
  v8f acc = {0.f, 0.f, 0.f, 0.f, 0.f, 0.f, 0.f, 0.f};
  for (int kc = 0; kc < 4; ++kc) {
    int k0 = kc * 64;
    for (int idx = tid; idx < 16 * 64; idx += 128) {
      int rr = idx >> 6, cc = idx & 63;
      size_t gi = (size_t)(m0 + rr) * DCH + k0 + cc;
      sA[idx] = yws[gi] + x[gi] * skipD[k0 + cc];
    }
    for (int idx = tid; idx < 64 * 64; idx += 128) {
      int kk = idx & 63, nn = idx >> 6;               // coalesced read of W row
      sW[kk * 65 + nn] = W[(size_t)(n0 + nn) * DCH + k0 + kk];
    }
    __syncthreads();
#pragma unroll
    for (int kb = 0; kb < 16; ++kb) {
      v2f a = fragA(sA, 64, 0, kb * 4, lane);
      v2f b = fragB(sW, 65, kb * 4, w * 16, lane);
      acc = WMMA_F32(a, b, acc);
    }
    __syncthreads();
  }
#pragma unroll
  for (int r = 0; r < 8; ++r) {
    int M = r + half * 8;
    out[(size_t)(m0 + M) * DCH + n0 + w * 16 + m] = acc[r] + sBias[w * 16 + m];
  }
}

extern "C" void kernel_launch(void* const* d_in, const int* in_sizes, int n_in,
                              void* d_out, int out_size, void* d_ws, size_t ws_size,
                              hipStream_t stream) {
  (void)in_sizes; (void)n_in; (void)out_size; (void)ws_size;
  const float* x         = (const float*)d_in[0];
  const float* logA      = (const float*)d_in[1];
  const float* Bp        = (const float*)d_in[2];
  const float* Cp        = (const float*)d_in[3];
  const float* log_delta = (const float*)d_in[4];
  const float* skipD     = (const float*)d_in[5];
  const float* W         = (const float*)d_in[6];
  const float* bout      = (const float*)d_in[7];
  float* out = (float*)d_out;
  float* ws  = (float*)d_ws;

  s4_prep_inv<<<1, 256, 0, stream>>>(logA, ws);
  s4_prep_X<<<DCH, NST, 0, stream>>>(Bp, ws + OFS_AINV, ws + OFS_X);
  s4_discretize<<<DCH, 256, 0, stream>>>(log_delta, ws + OFS_A, ws + OFS_X,
                                         ws + OFS_DA, ws + OFS_DB);
  s4_chunkops<<<DCH, 256, 0, stream>>>(Cp, ws + OFS_DA, ws + OFS_DB,
                                       ws + OFS_P, ws + OFS_E, ws + OFS_G, ws + OFS_KT);
  dim3 gT(BSZ * LSEQ / 32, DCH / 32), bT(32, 8);
  s4_transpose_x<<<gT, bT, 0, stream>>>(x, ws + OFS_XT);
  s4_scan<<<DCH, 32, 0, stream>>>(ws + OFS_XT, ws + OFS_P, ws + OFS_E,
                                  ws + OFS_G, ws + OFS_KT, ws + OFS_Y);
  dim3 gO(BSZ * LSEQ / 16, DCH / 64);
  s4_outgemm<<<gO, 128, 0, stream>>>(x, ws + OFS_Y, skipD, W, bout, out);
}